// AtomicConv_22720376995946
// MI455X (gfx1250) — compile-verified
//
#include <hip/hip_runtime.h>

#define TPB   256          // 8 wave32 per block
#define EPT   4            // edges per thread (one B128 per staged array)
#define TILE  (TPB * EPT)  // 1024 edges per tile
#define KF    16           // radial filters (reference K)
#define TF    4            // atom types (reference T)

// Low 32 bits of a flat shared-aperture pointer are the LDS byte offset
// (code-object ABI: AS3 -> generic keeps the offset in the low dword).
__device__ __forceinline__ unsigned lds_addr_of(const void* p) {
  return (unsigned)(unsigned long long)p;
}

// CDNA5 async copy engine: global -> LDS, 16B per lane, tracked by ASYNCcnt.
__device__ __forceinline__ void async_copy_b128(const void* gptr, unsigned lds) {
  unsigned long long ga = (unsigned long long)gptr;
  asm volatile("global_load_async_to_lds_b128 %0, %1, off"
               :: "v"(lds), "v"(ga)
               : "memory");
}

__global__ __launch_bounds__(TPB) void atomic_conv_kernel(
    const float* __restrict__ feat,      // (V,1) type ids as float
    const float* __restrict__ dist,      // (E,1)
    const int*   __restrict__ src,       // (E)
    const int*   __restrict__ dst,       // (E)
    const float* __restrict__ cutoffs,   // (K)
    const float* __restrict__ means,     // (K)
    const float* __restrict__ scaling,   // (K)
    const float* __restrict__ ftu,       // (T)
    float*       __restrict__ out,       // (V, T*K), pre-zeroed
    int E)
{
  __shared__ __align__(16) float sD[2][TILE];
  __shared__ __align__(16) int   sS[2][TILE];
  __shared__ __align__(16) int   sT[2][TILE];

  // Per-k constants held in registers (uniform scalar loads).
  // pk = pi/cutoff via v_rcp_f32 (1-ulp) instead of a ~10-op precise divide.
  float mk[KF], nvk[KF], ck[KF], pk[KF];
#pragma unroll
  for (int k = 0; k < KF; ++k) {
    mk[k]  = means[k];
    nvk[k] = -scaling[k];
    ck[k]  = cutoffs[k];
    pk[k]  = 3.14159265358979323846f * __builtin_amdgcn_rcpf(ck[k]);
  }
  const float f0 = ftu[0], f1 = ftu[1], f2 = ftu[2], f3 = ftu[3];

  const int tid = threadIdx.x;

  auto edge = [&](float d, int s, int dn) {
    const float fv = feat[s];
    int t;
    if      (fv == f0) t = 0;
    else if (fv == f1) t = 1;
    else if (fv == f2) t = 2;
    else if (fv == f3) t = 3;
    else return;                               // one-hot row all zero
    float* row = out + (long long)dn * (TF * KF) + t * KF;
#pragma unroll
    for (int k = 0; k < KF; ++k) {
      const float dm = d - mk[k];
      const float g  = __expf(nvk[k] * dm * dm);
      float c = 0.0f;
      if (d <= ck[k]) c = 0.5f * (__cosf(d * pk[k]) + 1.0f);
      const float v = g * c;
      // Non-returning f32 atomic add, agent scope (output is L2-resident).
      __hip_atomic_fetch_add(row + k, v, __ATOMIC_RELAXED,
                             __HIP_MEMORY_SCOPE_AGENT);
    }
  };

  const int fullTiles = E / TILE;
  const int t0 = blockIdx.x;
  int buf = 0;

  if (t0 < fullTiles) {                        // prime buffer 0
    const int base = t0 * TILE + tid * EPT;
    async_copy_b128(dist + base, lds_addr_of(&sD[0][tid * EPT]));
    async_copy_b128(src  + base, lds_addr_of(&sS[0][tid * EPT]));
    async_copy_b128(dst  + base, lds_addr_of(&sT[0][tid * EPT]));
  }

  for (int tile = t0; tile < fullTiles; tile += gridDim.x) {
    const int nxt = tile + gridDim.x;
    if (nxt < fullTiles) {                     // prefetch next tile, then wait
      const int nb   = nxt * TILE + tid * EPT; //   only for the current one
      const int pbuf = buf ^ 1;
      async_copy_b128(dist + nb, lds_addr_of(&sD[pbuf][tid * EPT]));
      async_copy_b128(src  + nb, lds_addr_of(&sS[pbuf][tid * EPT]));
      async_copy_b128(dst  + nb, lds_addr_of(&sT[pbuf][tid * EPT]));
      asm volatile("s_wait_asynccnt 3" ::: "memory");
    } else {
      asm volatile("s_wait_asynccnt 0" ::: "memory");
    }
#pragma unroll
    for (int j = 0; j < EPT; ++j) {
      edge(sD[buf][tid * EPT + j],
           sS[buf][tid * EPT + j],
           sT[buf][tid * EPT + j]);
    }
    buf ^= 1;
  }

  // Tail edges (E not a multiple of TILE): direct global loads.
  for (int e = fullTiles * TILE + blockIdx.x * TPB + tid; e < E;
       e += gridDim.x * TPB)
    edge(dist[e], src[e], dst[e]);
}

extern "C" void kernel_launch(void* const* d_in, const int* in_sizes, int n_in,
                              void* d_out, int out_size, void* d_ws, size_t ws_size,
                              hipStream_t stream) {
  const float* feat    = (const float*)d_in[0];
  const float* dist    = (const float*)d_in[1];
  const int*   src     = (const int*)d_in[2];
  const int*   dst     = (const int*)d_in[3];
  const float* cutoffs = (const float*)d_in[4];
  const float* means   = (const float*)d_in[5];
  const float* scaling = (const float*)d_in[6];
  const float* ftu     = (const float*)d_in[7];
  float*       out     = (float*)d_out;

  const int E = in_sizes[2];                   // edge count (src)

  // Output is accumulated with atomics: zero it first (graph-capturable).
  hipMemsetAsync(d_out, 0, sizeof(float) * (size_t)out_size, stream);

  const int fullTiles = E / TILE;
  int blocks = fullTiles > 0 ? fullTiles : 1;
  if (blocks > 1280) blocks = 1280;            // persistent grid, 2-3 tiles each

  atomic_conv_kernel<<<blocks, TPB, 0, stream>>>(
      feat, dist, src, dst, cutoffs, means, scaling, ftu, out, E);
}